// Linear_60876866453656
// MI455X (gfx1250) — compile-verified
//
#include <hip/hip_runtime.h>

typedef __attribute__((ext_vector_type(2))) float v2f;
typedef __attribute__((ext_vector_type(8))) float v8f;

static constexpr int kWaves    = 2;            // waves per block
static constexpr int kThreads  = kWaves * 32;  // wave32
static constexpr int kRowMax   = 64 * 5 + 4;   // padded LDS row (floats) for d=5
static constexpr int kTotalDim = 576;          // 64*(1+3+5)

// One irrep block: out[n, w, i] = 0.125 * sum_u x[n, u, i] * W[u, w]
// 16 samples per wave; WMMA f32 16x16x4 chained over u (K=4 * 16 steps).
template <int D>
__device__ __forceinline__ void irrep_block(
    const float* __restrict__ x, const float* __restrict__ W,
    float* __restrict__ out, float* __restrict__ xs,
    long n0, int off, int lane)
{
  const int half = lane >> 4;   // 0: lanes 0-15, 1: lanes 16-31
  const int lm   = lane & 15;
  constexpr int RS   = 64 * D + 4;   // padded LDS row stride (floats)
  constexpr int ROW4 = 16 * D;       // float4s per sample row
  constexpr int TOT4 = 16 * ROW4;    // float4s per 16-sample tile

  // Coalesced stage of x[n0..n0+15, off .. off+64*D) into wave-private LDS.
  #pragma unroll
  for (int f = lane; f < TOT4; f += 32) {
    const int r  = f / ROW4;
    const int c4 = f - r * ROW4;
    const float4 v = *(const float4*)(x + (n0 + r) * kTotalDim + off + c4 * 4);
    *(float4*)(xs + r * RS + c4 * 4) = v;
  }
  // Wave-private buffer + in-order DS ops => no workgroup barrier required.

  for (int wt = 0; wt < 4; ++wt) {
    const int wcol = wt * 16 + lm;
    // Preload B operands for all 16 K-steps (weights are L2/WGP$-resident).
    v2f b[16];
    #pragma unroll
    for (int kt = 0; kt < 16; ++kt) {
      const int u = kt * 4 + 2 * half;
      b[kt].x = 0.125f * W[u * 64 + wcol];        // fold sqrt(1/64)
      b[kt].y = 0.125f * W[(u + 1) * 64 + wcol];
    }
    #pragma unroll
    for (int i = 0; i < D; ++i) {
      v8f acc = {0.f, 0.f, 0.f, 0.f, 0.f, 0.f, 0.f, 0.f};
      #pragma unroll
      for (int kt = 0; kt < 16; ++kt) {
        const int u = kt * 4 + 2 * half;
        v2f a;
        a.x = xs[lm * RS + u * D + i];            // conflict-free banks
        a.y = xs[lm * RS + (u + 1) * D + i];
        acc = __builtin_amdgcn_wmma_f32_16x16x4_f32(
            false, a, false, b[kt], (short)0, acc, false, false);
      }
      // D-matrix layout: VGPR v -> row M = v + 8*half, col N = lm.
      #pragma unroll
      for (int v = 0; v < 8; ++v) {
        out[(n0 + v + 8 * half) * kTotalDim + off + wcol * D + i] = acc[v];
      }
    }
  }
}

__global__ void __launch_bounds__(kThreads)
linear_irreps_wmma(const float* __restrict__ x,
                   const float* __restrict__ w0,
                   const float* __restrict__ w1,
                   const float* __restrict__ w2,
                   float* __restrict__ out)
{
  __shared__ float lds_x[kWaves * 16 * kRowMax];  // ~41.5 KB
  const int lane = threadIdx.x & 31;
  const int wave = threadIdx.x >> 5;
  float* xs = lds_x + wave * 16 * kRowMax;
  const long n0 = ((long)blockIdx.x * kWaves + wave) * 16;

  irrep_block<1>(x, w0, out, xs, n0, 0,        lane);
  irrep_block<3>(x, w1, out, xs, n0, 64,       lane);
  irrep_block<5>(x, w2, out, xs, n0, 64 + 192, lane);
}

extern "C" void kernel_launch(void* const* d_in, const int* in_sizes, int n_in,
                              void* d_out, int out_size, void* d_ws, size_t ws_size,
                              hipStream_t stream) {
  const float* x  = (const float*)d_in[0];
  const float* w0 = (const float*)d_in[1];
  const float* w1 = (const float*)d_in[2];
  const float* w2 = (const float*)d_in[3];
  float* out = (float*)d_out;

  const int N = in_sizes[0] / kTotalDim;       // 262144
  const int blocks = N / (16 * kWaves);        // 8192, exact coverage
  linear_irreps_wmma<<<blocks, kThreads, 0, stream>>>(x, w0, w1, w2, out);
}